// CSGCNet_34376918237831
// MI455X (gfx1250) — compile-verified
//
#include <hip/hip_runtime.h>
#include <hip/hip_bf16.h>

#define BB   8
#define CC   64
#define HH   160
#define WW   160
#define LL   (HH * WW)          // 25600
#define CQ   8

typedef float v2f __attribute__((ext_vector_type(2)));
typedef float v8f __attribute__((ext_vector_type(8)));

__device__ __forceinline__ v8f wmma4(v2f a, v2f b, v8f c) {
    // D = A(16x4) * B(4x16) + C(16x16), fp32 WMMA on gfx1250
    return __builtin_amdgcn_wmma_f32_16x16x4_f32(false, a, false, b, (short)0, c,
                                                 false, false);
}

__device__ __forceinline__ float redmax16(float v) {
    v = fmaxf(v, __shfl_xor(v, 1, 32));
    v = fmaxf(v, __shfl_xor(v, 2, 32));
    v = fmaxf(v, __shfl_xor(v, 4, 32));
    v = fmaxf(v, __shfl_xor(v, 8, 32));
    return v;
}
__device__ __forceinline__ float redsum16(float v) {
    v += __shfl_xor(v, 1, 32);
    v += __shfl_xor(v, 2, 32);
    v += __shfl_xor(v, 4, 32);
    v += __shfl_xor(v, 8, 32);
    return v;
}

// ---------------------------------------------------------------------------
// Kernel 1: grouped 1x1 convs -> q, k, v in channel-last transposed layouts.
//   qT[b][h][w][8]            (= [b][l][8])
//   kW[b][h][j][8]            (j = w coordinate; = [b][l][8])
//   kH[b][w][j][8]            (j = h coordinate)
//   vW[b][h][j][64], vH[b][w][j][64]
// ---------------------------------------------------------------------------
__global__ void __launch_bounds__(256)
qkv_kernel(const float* __restrict__ x,
           const float* __restrict__ wq, const float* __restrict__ bq,
           const float* __restrict__ wk, const float* __restrict__ bk,
           const float* __restrict__ wv, const float* __restrict__ bv,
           float* __restrict__ qT, float* __restrict__ kH,
           float* __restrict__ kW, float* __restrict__ vH,
           float* __restrict__ vW)
{
    __shared__ float swq[CQ * 16], swk[CQ * 16], swv[CC * 16];
    __shared__ float sbq[CQ], sbk[CQ], sbv[CC];
    const int t = threadIdx.x;
    for (int i = t; i < CQ * 16; i += 256) { swq[i] = wq[i]; swk[i] = wk[i]; }
    for (int i = t; i < CC * 16; i += 256) swv[i] = wv[i];
    if (t < CQ) { sbq[t] = bq[t]; sbk[t] = bk[t]; }
    if (t < CC) sbv[t] = bv[t];
    __syncthreads();

    const int px = blockIdx.x * 256 + t;           // 0 .. B*L-1
    const int b  = px / LL;
    const int l  = px - b * LL;
    const int h  = l / WW;
    const int w  = l - h * WW;

    float xin[CC];
    const float* xb = x + (size_t)b * CC * LL + l;
#pragma unroll
    for (int c = 0; c < CC; ++c) xin[c] = xb[(size_t)c * LL];

    // q / k (2 outputs per group, 4 groups)
    float qv[CQ], kv[CQ];
#pragma unroll
    for (int cq = 0; cq < CQ; ++cq) {
        const int g = cq >> 1;
        float aq = sbq[cq], ak = sbk[cq];
#pragma unroll
        for (int ci = 0; ci < 16; ++ci) {
            const float xv = xin[g * 16 + ci];
            aq = fmaf(xv, swq[cq * 16 + ci], aq);
            ak = fmaf(xv, swk[cq * 16 + ci], ak);
        }
        qv[cq] = aq; kv[cq] = ak;
    }
    {
        float4* dq  = (float4*)(qT + ((size_t)b * LL + l) * CQ);
        float4* dkw = (float4*)(kW + ((size_t)b * LL + l) * CQ);
        float4* dkh = (float4*)(kH + (((size_t)b * WW + w) * HH + h) * CQ);
        dq[0]  = make_float4(qv[0], qv[1], qv[2], qv[3]);
        dq[1]  = make_float4(qv[4], qv[5], qv[6], qv[7]);
        dkw[0] = make_float4(kv[0], kv[1], kv[2], kv[3]);
        dkw[1] = make_float4(kv[4], kv[5], kv[6], kv[7]);
        dkh[0] = make_float4(kv[0], kv[1], kv[2], kv[3]);
        dkh[1] = make_float4(kv[4], kv[5], kv[6], kv[7]);
    }

    // v (16 outputs per group, 4 groups)
    float4* pW = (float4*)(vW + ((size_t)b * LL + l) * CC);
    float4* pH = (float4*)(vH + (((size_t)b * WW + w) * HH + h) * CC);
#pragma unroll
    for (int c4 = 0; c4 < 16; ++c4) {
        float acc[4];
#pragma unroll
        for (int j = 0; j < 4; ++j) {
            const int co = c4 * 4 + j;
            const int g  = co >> 4;
            float a = sbv[co];
#pragma unroll
            for (int ci = 0; ci < 16; ++ci)
                a = fmaf(xin[g * 16 + ci], swv[co * 16 + ci], a);
            acc[j] = a;
        }
        float4 r = make_float4(acc[0], acc[1], acc[2], acc[3]);
        pW[c4] = r;
        pH[c4] = r;
    }
}

// ---------------------------------------------------------------------------
// Kernel 2: fused criss-cross attention + grouped 1x1 conv (w1d, gamma folded).
// One workgroup = one (b, 16h x 16w) pixel tile; 8 waves; wave i owns
// columns {2i,2i+1} (H-attention) and rows {2i,2i+1} (W-attention).
// Energies kept in registers (v8f WMMA frags), softmax stats via LDS.
// Outputs o[b][l][64] (channel-last) plus per-WG/ per-wave BN stat partials.
// ---------------------------------------------------------------------------
__global__ void __launch_bounds__(256)
attn_kernel(const float* __restrict__ qT, const float* __restrict__ kH,
            const float* __restrict__ kW, const float* __restrict__ vH,
            const float* __restrict__ vW, const float* __restrict__ gamma,
            const float* __restrict__ w1d,
            float* __restrict__ o, float* __restrict__ partial)
{
    __shared__ float s_maxH[16][16], s_maxW[16][16];   // [h][w]
    __shared__ float s_sumH[16][16], s_sumW[16][16];
    __shared__ float s_att[8][16][17];                 // per-wave transpose buf
    __shared__ float s_tile[16][16][17];               // [hl][wl][c-chunk]
    __shared__ float s_w1[CC][16];                     // gamma * w1d

    const int t    = threadIdx.x;
    const int wave = t >> 5;
    const int lane = t & 31;
    const int ln   = lane & 15;   // N / M-in-16 index
    const int hi   = lane >> 4;   // lane half
    const int b    = blockIdx.z;
    const int ht0  = blockIdx.y * 16;
    const int wt0  = blockIdx.x * 16;

    {
        const float gm = gamma[0];
        for (int i = t; i < CC * 16; i += 256) s_w1[i >> 4][i & 15] = gm * w1d[i];
    }

    // ---- Q fragments (A operand, lane = M, VGPR pair = K {hi*2, hi*2+1}) ----
    v2f qAc[2][2];   // [col][kstep]  columns wl = 2*wave+cc, M = h
    v2f qAr[2][2];   // [row][kstep]  rows    hl = 2*wave+cc, M = w
#pragma unroll
    for (int cc = 0; cc < 2; ++cc) {
#pragma unroll
        for (int kk = 0; kk < 2; ++kk) {
            const int hg = ht0 + ln, wg = wt0 + 2 * wave + cc;
            qAc[cc][kk] = *(const v2f*)(qT + ((size_t)(b * LL + hg * WW + wg)) * CQ
                                        + kk * 4 + hi * 2);
            const int hg2 = ht0 + 2 * wave + cc, wg2 = wt0 + ln;
            qAr[cc][kk] = *(const v2f*)(qT + ((size_t)(b * LL + hg2 * WW + wg2)) * CQ
                                        + kk * 4 + hi * 2);
        }
    }

    // ---- Phase 1: energies (kept in registers), per-pixel maxima ----------
    v8f EH[2][10], EW[2][10];
#pragma unroll
    for (int cc = 0; cc < 2; ++cc) {
        const int wl = 2 * wave + cc;
        const float* kb = kH + ((size_t)(b * WW + wt0 + wl)) * HH * CQ;  // [j][8]
        float rm[8];
#pragma unroll
        for (int r = 0; r < 8; ++r) rm[r] = -3.0e38f;
        for (int jt = 0; jt < 10; ++jt) {
            const int jg = jt * 16 + ln;
            v2f b0 = *(const v2f*)(kb + jg * CQ + hi * 2);
            v2f b1 = *(const v2f*)(kb + jg * CQ + 4 + hi * 2);
            v8f e = {};
            e = wmma4(qAc[cc][0], b0, e);
            e = wmma4(qAc[cc][1], b1, e);
#pragma unroll
            for (int r = 0; r < 8; ++r) {           // diagonal mask (j == h)
                if (jt * 16 + ln == ht0 + r + hi * 8) e[r] = -1.0e30f;
                rm[r] = fmaxf(rm[r], e[r]);
            }
            EH[cc][jt] = e;
        }
#pragma unroll
        for (int r = 0; r < 8; ++r) {
            const float m = redmax16(rm[r]);
            if (ln == 0) s_maxH[r + hi * 8][wl] = m;
        }
    }
#pragma unroll
    for (int cc = 0; cc < 2; ++cc) {
        const int hl = 2 * wave + cc;
        const float* kb = kW + ((size_t)(b * HH + ht0 + hl)) * WW * CQ;  // [j][8]
        float rm[8];
#pragma unroll
        for (int r = 0; r < 8; ++r) rm[r] = -3.0e38f;
        for (int jt = 0; jt < 10; ++jt) {
            const int jg = jt * 16 + ln;
            v2f b0 = *(const v2f*)(kb + jg * CQ + hi * 2);
            v2f b1 = *(const v2f*)(kb + jg * CQ + 4 + hi * 2);
            v8f e = {};
            e = wmma4(qAr[cc][0], b0, e);
            e = wmma4(qAr[cc][1], b1, e);
#pragma unroll
            for (int r = 0; r < 8; ++r) rm[r] = fmaxf(rm[r], e[r]);
            EW[cc][jt] = e;
        }
#pragma unroll
        for (int r = 0; r < 8; ++r) {
            const float m = redmax16(rm[r]);
            if (ln == 0) s_maxW[hl][r + hi * 8] = m;
        }
    }
    __syncthreads();

    // ---- Phase 2: exponentials (in place) + per-pixel partial sums --------
#pragma unroll
    for (int cc = 0; cc < 2; ++cc) {
        const int wl = 2 * wave + cc;
        float Mv[8], rs[8];
#pragma unroll
        for (int r = 0; r < 8; ++r) {
            const int m = r + hi * 8;
            Mv[r] = fmaxf(s_maxH[m][wl], s_maxW[m][wl]);
            rs[r] = 0.f;
        }
        for (int jt = 0; jt < 10; ++jt) {
            v8f e = EH[cc][jt];
#pragma unroll
            for (int r = 0; r < 8; ++r) {
                const float ex = __expf(e[r] - Mv[r]);
                e[r] = ex; rs[r] += ex;
            }
            EH[cc][jt] = e;
        }
#pragma unroll
        for (int r = 0; r < 8; ++r) {
            const float s = redsum16(rs[r]);
            if (ln == 0) s_sumH[r + hi * 8][wl] = s;
        }
    }
#pragma unroll
    for (int cc = 0; cc < 2; ++cc) {
        const int hl = 2 * wave + cc;
        float Mv[8], rs[8];
#pragma unroll
        for (int r = 0; r < 8; ++r) {
            const int m = r + hi * 8;
            Mv[r] = fmaxf(s_maxH[hl][m], s_maxW[hl][m]);
            rs[r] = 0.f;
        }
        for (int jt = 0; jt < 10; ++jt) {
            v8f e = EW[cc][jt];
#pragma unroll
            for (int r = 0; r < 8; ++r) {
                const float ex = __expf(e[r] - Mv[r]);
                e[r] = ex; rs[r] += ex;
            }
            EW[cc][jt] = e;
        }
#pragma unroll
        for (int r = 0; r < 8; ++r) {
            const float s = redsum16(rs[r]);
            if (ln == 0) s_sumW[hl][r + hi * 8] = s;
        }
    }
    __syncthreads();

    // reciprocal softmax denominators for owned pixels
    float rdenH[2][8], rdenW[2][8];
#pragma unroll
    for (int cc = 0; cc < 2; ++cc) {
#pragma unroll
        for (int r = 0; r < 8; ++r) {
            const int m = r + hi * 8;
            rdenH[cc][r] = 1.f / (s_sumH[m][2 * wave + cc] + s_sumW[m][2 * wave + cc]);
            rdenW[cc][r] = 1.f / (s_sumH[2 * wave + cc][m] + s_sumW[2 * wave + cc][m]);
        }
    }

    // ---- Phase 3/4: aggregation + grouped conv, in 16-channel chunks ------
    float accS[4], accQ[4];   // BN partial sums, channel = cb*16 + ln
#pragma unroll
    for (int cb = 0; cb < 4; ++cb) { accS[cb] = 0.f; accQ[cb] = 0.f; }

    for (int cb = 0; cb < 4; ++cb) {
        for (int i = t; i < 16 * 16 * 16; i += 256)
            s_tile[i >> 8][(i >> 4) & 15][i & 15] = 0.f;
        __syncthreads();

        // out_H : columns
#pragma unroll
        for (int cc = 0; cc < 2; ++cc) {
            const int wl = 2 * wave + cc;
            const float* vb = vH + ((size_t)(b * WW + wt0 + wl)) * HH * CC + cb * 16;
            v8f acc = {};
            for (int jt = 0; jt < 10; ++jt) {
                v8f e = EH[cc][jt];
#pragma unroll
                for (int r = 0; r < 8; ++r)
                    s_att[wave][r + hi * 8][ln] = e[r] * rdenH[cc][r];
#pragma unroll
                for (int kk = 0; kk < 4; ++kk) {
                    const int k0 = kk * 4 + hi * 2;
                    v2f a; a.x = s_att[wave][ln][k0]; a.y = s_att[wave][ln][k0 + 1];
                    const int j0 = jt * 16 + k0;
                    v2f bf; bf.x = vb[(size_t)j0 * CC + ln];
                    bf.y = vb[(size_t)(j0 + 1) * CC + ln];
                    acc = wmma4(a, bf, acc);
                }
            }
#pragma unroll
            for (int r = 0; r < 8; ++r) s_tile[r + hi * 8][wl][ln] = acc[r];
        }
        __syncthreads();

        // out_W : rows (accumulate)
#pragma unroll
        for (int cc = 0; cc < 2; ++cc) {
            const int hl = 2 * wave + cc;
            const float* vb = vW + ((size_t)(b * HH + ht0 + hl)) * WW * CC + cb * 16;
            v8f acc = {};
            for (int jt = 0; jt < 10; ++jt) {
                v8f e = EW[cc][jt];
#pragma unroll
                for (int r = 0; r < 8; ++r)
                    s_att[wave][r + hi * 8][ln] = e[r] * rdenW[cc][r];
#pragma unroll
                for (int kk = 0; kk < 4; ++kk) {
                    const int k0 = kk * 4 + hi * 2;
                    v2f a; a.x = s_att[wave][ln][k0]; a.y = s_att[wave][ln][k0 + 1];
                    const int j0 = jt * 16 + k0;
                    v2f bf; bf.x = vb[(size_t)j0 * CC + ln];
                    bf.y = vb[(size_t)(j0 + 1) * CC + ln];
                    acc = wmma4(a, bf, acc);
                }
            }
#pragma unroll
            for (int r = 0; r < 8; ++r) s_tile[hl][r + hi * 8][ln] += acc[r];
        }
        __syncthreads();

        // grouped conv w1d (group g == cb; gamma pre-folded), write o chunk
#pragma unroll
        for (int cc = 0; cc < 2; ++cc) {
            const int hl = 2 * wave + cc;
            v8f acc = {};
#pragma unroll
            for (int kk = 0; kk < 4; ++kk) {
                const int k0 = kk * 4 + hi * 2;
                v2f a; a.x = s_tile[hl][ln][k0]; a.y = s_tile[hl][ln][k0 + 1];
                v2f bf; bf.x = s_w1[cb * 16 + ln][k0]; bf.y = s_w1[cb * 16 + ln][k0 + 1];
                acc = wmma4(a, bf, acc);
            }
            float s = 0.f, s2 = 0.f;
#pragma unroll
            for (int r = 0; r < 8; ++r) {
                const float v = acc[r];
                const int l = (ht0 + hl) * WW + wt0 + r + hi * 8;
                o[((size_t)b * LL + l) * CC + cb * 16 + ln] = v;
                s += v; s2 += v * v;
            }
            s  += __shfl_xor(s, 16, 32);
            s2 += __shfl_xor(s2, 16, 32);
            accS[cb] += s; accQ[cb] += s2;
        }
        __syncthreads();
    }

    // deterministic BN stat partials: [wg][wave][128]
    if (lane < 16) {
        const int wg = (blockIdx.z * gridDim.y + blockIdx.y) * gridDim.x + blockIdx.x;
        float* p = partial + ((size_t)wg * 8 + wave) * 128;
#pragma unroll
        for (int cb = 0; cb < 4; ++cb) {
            p[cb * 16 + ln]      = accS[cb];
            p[64 + cb * 16 + ln] = accQ[cb];
        }
    }
}

// ---------------------------------------------------------------------------
// Kernel 3: deterministic reduction of BN partials -> gsums[128]
// ---------------------------------------------------------------------------
__global__ void __launch_bounds__(128)
stat_reduce_kernel(const float* __restrict__ partial, float* __restrict__ gsums,
                   int nslices)
{
    const int t = threadIdx.x;   // 0..127
    float s = 0.f;
    for (int j = 0; j < nslices; ++j) s += partial[(size_t)j * 128 + t];
    gsums[t] = s;
}

// ---------------------------------------------------------------------------
// Kernel 4: BatchNorm + residual + ReLU, LDS transpose for coalescing.
// ---------------------------------------------------------------------------
__global__ void __launch_bounds__(256)
bn_out_kernel(const float* __restrict__ o, const float* __restrict__ x,
              const float* __restrict__ gsums, const float* __restrict__ bn_w,
              const float* __restrict__ bn_b, float* __restrict__ out)
{
    __shared__ float tile[64][65];
    __shared__ float sscale[64], sshift[64];
    const int t  = threadIdx.x;
    const int b  = blockIdx.y;
    const int l0 = blockIdx.x * 64;
    if (t < 64) {
        const float inv  = 1.f / ((float)BB * (float)LL);
        const float mean = gsums[t] * inv;
        const float var  = gsums[64 + t] * inv - mean * mean;
        const float rs   = rsqrtf(var + 1e-5f);
        sscale[t] = bn_w[t] * rs;
        sshift[t] = bn_b[t] - mean * bn_w[t] * rs;
    }
    const float* ob = o + ((size_t)b * LL + l0) * CC;
    for (int i = t; i < 64 * 64; i += 256) tile[i >> 6][i & 63] = ob[i];
    __syncthreads();
    for (int i = t; i < 64 * 64; i += 256) {
        const int c = i >> 6, l = i & 63;
        const size_t idx = ((size_t)b * CC + c) * LL + l0 + l;
        const float v = tile[l][c] * sscale[c] + sshift[c] + x[idx];
        out[idx] = fmaxf(v, 0.f);
    }
}

// ---------------------------------------------------------------------------
extern "C" void kernel_launch(void* const* d_in, const int* in_sizes, int n_in,
                              void* d_out, int out_size, void* d_ws, size_t ws_size,
                              hipStream_t stream) {
    const float* x     = (const float*)d_in[0];
    const float* wq    = (const float*)d_in[1];
    const float* bq    = (const float*)d_in[2];
    const float* wk    = (const float*)d_in[3];
    const float* bk    = (const float*)d_in[4];
    const float* wv    = (const float*)d_in[5];
    const float* bv    = (const float*)d_in[6];
    const float* gamma = (const float*)d_in[7];
    const float* w1d   = (const float*)d_in[8];
    const float* bn_w  = (const float*)d_in[9];
    const float* bn_b  = (const float*)d_in[10];
    float* out = (float*)d_out;

    // workspace layout (floats)
    float* qT  = (float*)d_ws;                          // B*L*8
    float* kHs = qT  + (size_t)BB * LL * CQ;            // B*L*8
    float* kWs = kHs + (size_t)BB * LL * CQ;            // B*L*8
    float* vHs = kWs + (size_t)BB * LL * CQ;            // B*L*64
    float* vWs = vHs + (size_t)BB * LL * CC;            // B*L*64
    float* o   = vWs + (size_t)BB * LL * CC;            // B*L*64
    float* prt = o   + (size_t)BB * LL * CC;            // 800*8*128
    float* gsums = prt + (size_t)800 * 8 * 128;         // 128

    qkv_kernel<<<dim3((BB * LL) / 256), dim3(256), 0, stream>>>(
        x, wq, bq, wk, bk, wv, bv, qT, kHs, kWs, vHs, vWs);

    attn_kernel<<<dim3(WW / 16, HH / 16, BB), dim3(256), 0, stream>>>(
        qT, kHs, kWs, vHs, vWs, gamma, w1d, o, prt);

    stat_reduce_kernel<<<dim3(1), dim3(128), 0, stream>>>(prt, gsums, 800 * 8);

    bn_out_kernel<<<dim3(LL / 64, BB), dim3(256), 0, stream>>>(
        o, x, gsums, bn_w, bn_b, out);
}